// DeepAutoreg_27109833572726
// MI455X (gfx1250) — compile-verified
//
#include <hip/hip_runtime.h>
#include <stdint.h>

typedef __attribute__((ext_vector_type(2))) float        v2f;
typedef __attribute__((ext_vector_type(8))) float        v8f;
typedef __attribute__((ext_vector_type(4))) unsigned int v4u;
typedef __attribute__((ext_vector_type(4))) int          v4i;
typedef __attribute__((ext_vector_type(8))) int          v8i;

#define NEG_SLOPE 0.01f
#define D       128
#define NSRC    2048
#define NNODES  20000
#define XPAD    132            // mid-tile row stride (dwords), conflict-free
#define GROW    130            // kernel1 gathered-x padded row stride (TDM pad: 2 dw / 128 dw)
#define KHALF   1024           // A columns per TDM descriptor (kernel 2)
#define HROW    1028           // padded A row stride in dwords (TDM pad: 1 dw / 256 dw)
#define HSTRM   (16 * HROW)    // dwords per TDM half-stream (16448)

// ---------------- TDM descriptor helper (2D tile, 4-byte elements) ----------------
// g0: count=1 | gather flags, lds byte addr, 57-bit global addr, type=2
// g1: data_size=4B, pad cfg, tensor_dim0/1, tile_dim0/1, tensor_dim0_stride
__device__ __forceinline__ void tdm_load_2d(
    uint32_t lds_addr, uint64_t gaddr,
    uint32_t tensor_d0, uint32_t tensor_d1, uint32_t tile_d0, uint32_t tile_d1,
    uint32_t row_stride, uint32_t pad_interval_code, uint32_t pad_amount_code,
    uint32_t gather, v4i g2, v4i g3)
{
    v4u g0;
    g0[0] = 1u | (gather << 31);                       // count=1, gather_mode, 16-bit idx
    g0[1] = lds_addr;
    g0[2] = (uint32_t)gaddr;
    g0[3] = ((uint32_t)(gaddr >> 32) & 0x01FFFFFFu) | 0x80000000u;   // type=2 ("image")
    v8i g1;
    g1[0] = (int)((2u << 16) | (1u << 20) |            // data_size=4B, pad_enable
                  (pad_interval_code << 22) | (pad_amount_code << 25));
    g1[1] = (int)((tensor_d0 & 0xFFFFu) << 16);        // tensor_dim0[15:0] at bits 63:48
    g1[2] = (int)((tensor_d0 >> 16) | ((tensor_d1 & 0xFFFFu) << 16));
    g1[3] = (int)((tensor_d1 >> 16) | (tile_d0 << 16));
    g1[4] = (int)(tile_d1 & 0xFFFFu);                  // tile_dim1 (tile_dim2 = 0)
    g1[5] = (int)row_stride;                           // tensor_dim0_stride[31:0]
    g1[6] = 0;
    g1[7] = 0;
#if __clang_major__ >= 23
    v8i z8 = {0, 0, 0, 0, 0, 0, 0, 0};
    __builtin_amdgcn_tensor_load_to_lds(g0, g1, g2, g3, z8, 0);
#else
    __builtin_amdgcn_tensor_load_to_lds(g0, g1, g2, g3, 0);
#endif
}

// -------- Kernel 1: support[s][n] = sum_k x[src_idx[s]][k] * W[k][n] --------
// TDM gather-mode DMA pulls 16 x-rows (sorted 16-bit indices) into padded LDS.
__global__ __launch_bounds__(256) void gcn_gather_transform(
    const float* __restrict__ x,
    const float* __restrict__ W,
    const long long* __restrict__ src_idx,
    float* __restrict__ support)
{
    __shared__ float ldsX[16 * GROW];       // 8320 B, stride 130 => banks (2r+c)%64
    const int tid  = threadIdx.x;
    const int lane = tid & 31;
    const int wave = tid >> 5;
    const int m0   = blockIdx.x * 16;

    if (wave == 0) {
        // pack 16 sorted row indices (all < 65536) as 16-bit gather indices
        unsigned iw[8];
        #pragma unroll
        for (int p = 0; p < 8; ++p) {
            unsigned lo = (unsigned)src_idx[m0 + 2 * p];
            unsigned hi = (unsigned)src_idx[m0 + 2 * p + 1];
            iw[p] = (lo & 0xFFFFu) | (hi << 16);
        }
        v4i g2 = {(int)iw[0], (int)iw[1], (int)iw[2], (int)iw[3]};
        v4i g3 = {(int)iw[4], (int)iw[5], (int)iw[6], (int)iw[7]};
        tdm_load_2d((uint32_t)(uintptr_t)(&ldsX[0]),
                    (uint64_t)(uintptr_t)x,
                    /*tensor_d0=*/D, /*tensor_d1=*/NNODES,
                    /*tile_d0=*/D,   /*tile_d1(=#indices)=*/16,
                    /*row_stride=*/D,
                    /*pad_interval (6 -> 128 dw)=*/6u, /*pad_amount (1 -> 2 dw)=*/1u,
                    /*gather=*/1u, g2, g3);
        __builtin_amdgcn_s_wait_tensorcnt(0);
    }
    __syncthreads();

    const int n0   = wave * 16;
    const int mrow = lane & 15;             // A-operand M row / B-operand N col
    const int koff = (lane >> 4) * 2;       // K sub-offset per half-wave

    v8f c = {};
    #pragma unroll 4
    for (int k = 0; k < D; k += 4) {
        const int kl = k + koff;
        v2f a, b;
        a[0] = ldsX[mrow * GROW + kl];
        a[1] = ldsX[mrow * GROW + kl + 1];
        b[0] = W[(size_t)kl * D + n0 + mrow];
        b[1] = W[(size_t)(kl + 1) * D + n0 + mrow];
        c = __builtin_amdgcn_wmma_f32_16x16x4_f32(false, a, false, b,
                                                  (short)0, c, false, false);
    }

    const int mbase = m0 + ((lane >> 4) << 3);
    const int nout  = n0 + mrow;
    #pragma unroll
    for (int v = 0; v < 8; ++v)
        support[(size_t)(mbase + v) * D + nout] = c[v];
}

// ---- Kernel 2: out = LeakyReLU(A @ support) @ dense0_w^T, fused per block ----
// Two TDM descriptors DMA the full 16x2048 A row-block (padded, bank-conflict-free);
// compute on half 0 overlaps the DMA of half 1. Only 3 barriers per block.
__global__ __launch_bounds__(256) void gcn_aggregate_dense(
    const float* __restrict__ A,        // [NNODES][NSRC]
    const float* __restrict__ support,  // [NSRC][D]  (L2-resident, 1 MB)
    const float* __restrict__ dw,       // dense0_w [D][D] row-major (o, k)
    float* __restrict__ out)            // [NNODES][D]
{
    __shared__ float ldsA[2 * HSTRM];   // 131584 B: padded stride 1028 => banks (4r+c)%64
    __shared__ float ldsM[16 * XPAD];   // 8448 B mid tile

    const int tid  = threadIdx.x;
    const int lane = tid & 31;
    const int wave = tid >> 5;
    const int m0   = blockIdx.x * 16;
    const int n0   = wave * 16;
    const int mrow = lane & 15;
    const int koff = (lane >> 4) * 2;

    if (wave == 0) {
        const uint64_t gaBase  = (uint64_t)(uintptr_t)(A + (size_t)m0 * NSRC);
        const uint32_t ldsBase = (uint32_t)(uintptr_t)(&ldsA[0]);
        v4i z = {0, 0, 0, 0};
        #pragma unroll
        for (int h = 0; h < 2; ++h)
            tdm_load_2d(ldsBase + (uint32_t)h * (HSTRM * 4),
                        gaBase + (uint64_t)h * (KHALF * 4),
                        /*tensor_d0=*/KHALF, /*tensor_d1=*/16,
                        /*tile_d0=*/KHALF,   /*tile_d1=*/16,
                        /*row_stride=*/NSRC,
                        /*pad_interval (7 -> 256 dw)=*/7u, /*pad_amount (0 -> 1 dw)=*/0u,
                        /*gather=*/0u, z, z);
    }

    v8f c = {};
    #pragma unroll
    for (int h = 0; h < 2; ++h) {
        if (wave == 0) {
            if (h == 0) __builtin_amdgcn_s_wait_tensorcnt(1);  // half 0 landed, #2 in flight
            else        __builtin_amdgcn_s_wait_tensorcnt(0);  // all DMA done
        }
        __syncthreads();

        const float* hs = ldsA + h * HSTRM + mrow * HROW;
        for (int kb = 0; kb < KHALF; kb += 256) {
            const float* lb = hs + kb + (kb >> 8);      // +1 dw pad per 256 dw
            const float* sb = support + (size_t)(h * KHALF + kb) * D + n0 + mrow;
            #pragma unroll 8
            for (int kk = 0; kk < 256; kk += 4) {
                const int kl = kk + koff;
                v2f a, b;
                a[0] = lb[kl];
                a[1] = lb[kl + 1];
                b[0] = sb[(size_t)kl * D];
                b[1] = sb[(size_t)(kl + 1) * D];
                c = __builtin_amdgcn_wmma_f32_16x16x4_f32(false, a, false, b,
                                                          (short)0, c, false, false);
            }
        }
    }

    // LeakyReLU in-register, stage mid tile [16][128] to LDS
    const int mhalf = (lane >> 4) << 3;     // 0 or 8
    #pragma unroll
    for (int v = 0; v < 8; ++v) {
        float m = c[v];
        m = (m >= 0.0f) ? m : NEG_SLOPE * m;
        ldsM[(mhalf + v) * XPAD + n0 + mrow] = m;
    }
    __syncthreads();

    // out[m][o] = sum_k mid[m][k] * dw[o][k]; wave covers o0 = 16*wave, K = 128
    v8f c2 = {};
    const int o0 = n0;
    #pragma unroll 4
    for (int k = 0; k < D; k += 4) {
        const int kl = k + koff;
        v2f a, b;
        a[0] = ldsM[mrow * XPAD + kl];
        a[1] = ldsM[mrow * XPAD + kl + 1];
        b[0] = dw[(size_t)(o0 + mrow) * D + kl];      // B[k][o] = dw[o][k]
        b[1] = dw[(size_t)(o0 + mrow) * D + kl + 1];
        c2 = __builtin_amdgcn_wmma_f32_16x16x4_f32(false, a, false, b,
                                                   (short)0, c2, false, false);
    }

    const int mbase = m0 + mhalf;
    const int oout  = o0 + mrow;
    #pragma unroll
    for (int v = 0; v < 8; ++v)
        out[(size_t)(mbase + v) * D + oout] = c2[v];
}

extern "C" void kernel_launch(void* const* d_in, const int* in_sizes, int n_in,
                              void* d_out, int out_size, void* d_ws, size_t ws_size,
                              hipStream_t stream) {
    (void)in_sizes; (void)n_in; (void)out_size; (void)ws_size;
    const float*     x   = (const float*)d_in[0];
    const float*     A   = (const float*)d_in[1];
    const float*     W   = (const float*)d_in[2];
    const float*     dw  = (const float*)d_in[3];
    const long long* idx = (const long long*)d_in[4];   // src_idx is int64
    float* out     = (float*)d_out;
    float* support = (float*)d_ws;                      // NSRC*D*4 = 1 MB scratch

    gcn_gather_transform<<<NSRC / 16, 256, 0, stream>>>(x, W, idx, support);
    gcn_aggregate_dense<<<NNODES / 16, 256, 0, stream>>>(A, support, dw, out);
}